// ImprovedClusteredAttention_73443940761952
// MI455X (gfx1250) — compile-verified
//
#include <hip/hip_runtime.h>
#include <hip/hip_bf16.h>
#include <math.h>

// Problem constants (from reference)
#define NN 2
#define LL 2048
#define SS 2048
#define HH 8
#define EE 64
#define DD 64
#define CC 256
#define KK 32
#define TEMP 0.125f          // 1/sqrt(64)
#define NEG_LARGE (-3.0e38f)
#define SPLIT 4              // K-dim split for the P*V GEMM
#define SCHUNK (SS / SPLIT)  // 512

typedef float v2f __attribute__((ext_vector_type(2)));
typedef float v8f __attribute__((ext_vector_type(8)));

// workspace element counts
#define NG   (NN*HH*CC*EE)         // Qg        262144 f
#define NHC  (NN*HH*CC)            //           4096 f
#define NP   ((size_t)NN*HH*CC*SS) // scores/probs 8388608 f
#define NTK  (NN*HH*CC*KK)         // topk idx  131072 i
#define NVB  (NN*HH*CC*DD)         // V_btm_c   262144 f

// ---------------------------------------------------------------- zero init
__global__ void zero_f32(float* p, size_t n) {
  size_t i = (size_t)blockIdx.x * blockDim.x + threadIdx.x;
  if (i < n) p[i] = 0.0f;
}

// ------------------------------------------------- cluster sums via atomics
// queries [N,L,H,E], cid [N,H,L] -> Qsum [N,H,C,E], counts [N,H,C]
__global__ void cluster_accum(const float* __restrict__ queries,
                              const int* __restrict__ cid,
                              float* __restrict__ Qsum,
                              float* __restrict__ counts) {
  size_t i = (size_t)blockIdx.x * blockDim.x + threadIdx.x; // over N*H*L
  if (i >= (size_t)NN * HH * LL) return;
  int l  = (int)(i % LL);
  int nh = (int)(i / LL);
  int h  = nh % HH, n = nh / HH;
  int c  = cid[i];                                   // cid flat = nh*L + l = i
  const float* q = queries + (((size_t)n * LL + l) * HH + h) * EE;
  float* dst = Qsum + ((size_t)nh * CC + c) * EE;
  for (int e = 0; e < EE; ++e) atomicAdd(&dst[e], q[e]);
  atomicAdd(&counts[(size_t)nh * CC + c], 1.0f);
}

// Qg = Qsum / max(count,1)
__global__ void finalize_means(float* __restrict__ Qg,
                               const float* __restrict__ counts) {
  size_t i = (size_t)blockIdx.x * blockDim.x + threadIdx.x;
  if (i >= (size_t)NG) return;
  float cnt = counts[i / EE];
  Qg[i] = Qg[i] / fmaxf(cnt, 1.0f);
}

// ------------------------------------------------- WMMA GEMM: QK = Qg * K^T
// Qg [NH,C,E], keys [N,S,H,E] -> P [NH,C,S]   (raw scores, temp applied later)
// One wave computes a 16(M) x 64(N) strip: A fragment reused across 4 WMMAs.
__global__ void qk_gemm(const float* __restrict__ Qg,
                        const float* __restrict__ keys,
                        float* __restrict__ P) {
  const int SG = SS / 64;                            // 32 N-strips per head
  const int TPH = (CC / 16) * SG;                    // 512 strips per head
  int wid = blockIdx.x * (blockDim.x >> 5) + (threadIdx.x >> 5);
  int nh = wid / TPH;
  int t  = wid % TPH;
  int mt = t / SG;                                   // cluster tile (0..15)
  int nt = t % SG;                                   // 64-wide key strip
  int n = nh / HH, h = nh % HH;
  int lane = threadIdx.x & 31;
  int half = lane >> 4, tl = lane & 15;
  int m0 = mt * 16, s0 = nt * 64;

  const float* qrow = Qg + ((size_t)nh * CC + m0 + tl) * EE;
  const size_t kstride = (size_t)HH * EE;            // between key rows
  const float* krow0 = keys + (((size_t)n * SS + s0 + tl) * HH + h) * EE;

  v8f acc0 = {}, acc1 = {}, acc2 = {}, acc3 = {};
  for (int e = 0; e < EE; e += 4) {
    v2f a, b0, b1, b2, b3;
    // A(16x4): lanes 0-15 -> K = e+0,e+1 ; lanes 16-31 -> K = e+2,e+3
    a.x = qrow[e + 2 * half];
    a.y = qrow[e + 2 * half + 1];
    // B(4x16) for 4 adjacent 16-key tiles
    const float* k0 = krow0 + e + 2 * half;
    b0.x = k0[0];                  b0.y = k0[1];
    b1.x = k0[16 * kstride];       b1.y = k0[16 * kstride + 1];
    b2.x = k0[32 * kstride];       b2.y = k0[32 * kstride + 1];
    b3.x = k0[48 * kstride];       b3.y = k0[48 * kstride + 1];
    acc0 = __builtin_amdgcn_wmma_f32_16x16x4_f32(false, a, false, b0, (short)0, acc0, false, false);
    acc1 = __builtin_amdgcn_wmma_f32_16x16x4_f32(false, a, false, b1, (short)0, acc1, false, false);
    acc2 = __builtin_amdgcn_wmma_f32_16x16x4_f32(false, a, false, b2, (short)0, acc2, false, false);
    acc3 = __builtin_amdgcn_wmma_f32_16x16x4_f32(false, a, false, b3, (short)0, acc3, false, false);
  }
  // D layout: VGPR r -> row m0 + r + 8*half, col (s-tile) + tl
  float* prow = P + ((size_t)nh * CC + m0 + 8 * half) * SS + s0 + tl;
#pragma unroll
  for (int r = 0; r < 8; ++r) {
    prow[(size_t)r * SS +  0] = acc0[r];
    prow[(size_t)r * SS + 16] = acc1[r];
    prow[(size_t)r * SS + 32] = acc2[r];
    prow[(size_t)r * SS + 48] = acc3[r];
  }
}

// ------------------------- per-row top-K selection + masked softmax in LDS
// P row [S] -> topk idx [K], Abtm scalar, P row overwritten with bottom-k probs
__global__ void topk_softmax(float* __restrict__ P,
                             int* __restrict__ topk,
                             float* __restrict__ Abtm) {
  __shared__ float vals[SS];        // 8 KB of the 320 KB WGP LDS
  __shared__ float swv[8];
  __shared__ int   swi[8];
  int row = blockIdx.x;             // 0 .. N*H*C-1
  int tid = threadIdx.x;
  int lane = tid & 31, wv = tid >> 5;
  float* prow = P + (size_t)row * SS;

  for (int s = tid; s < SS; s += 256) vals[s] = prow[s];
  __syncthreads();

  float m = 0.0f;                   // row max (set at round 0)
  float topk_expsum = 0.0f;
  for (int r = 0; r < KK; ++r) {
    // local argmax over this thread's 8 strided elements (lowest index wins)
    float lm = NEG_LARGE; int li = 0;
    for (int s = tid; s < SS; s += 256) {
      float v = vals[s];
      if (v > lm) { lm = v; li = s; }
    }
    // wave32 argmax reduce with lowest-index tie-break
    for (int off = 16; off > 0; off >>= 1) {
      float ov = __shfl_xor(lm, off);
      int   oi = __shfl_xor(li, off);
      if (ov > lm || (ov == lm && oi < li)) { lm = ov; li = oi; }
    }
    if (lane == 0) { swv[wv] = lm; swi[wv] = li; }
    __syncthreads();
    if (tid < 32) {                 // wave 0 reduces the 8 per-wave winners
      float fv = (tid < 8) ? swv[tid] : NEG_LARGE;
      int   fi = (tid < 8) ? swi[tid] : 0;
      for (int off = 4; off > 0; off >>= 1) {
        float ov = __shfl_xor(fv, off);
        int   oi = __shfl_xor(fi, off);
        if (ov > fv || (ov == fv && oi < fi)) { fv = ov; fi = oi; }
      }
      if (tid == 0) {
        swv[0] = fv; swi[0] = fi;
        vals[fi] = NEG_LARGE;       // exclude from further rounds
        topk[(size_t)row * KK + r] = fi;
      }
    }
    __syncthreads();
    float gv = swv[0];
    if (r == 0) m = gv;
    topk_expsum += __expf(TEMP * (gv - m));
    __syncthreads();                // protect swv[0] before next round's writes
  }

  // bottom-k: exp of remaining entries (selected ones -> 0)
  float lsum = 0.0f;
  for (int s = tid; s < SS; s += 256) {
    float e = __expf(TEMP * (vals[s] - m));
    vals[s] = e;
    lsum += e;
  }
  for (int off = 16; off > 0; off >>= 1) lsum += __shfl_xor(lsum, off);
  if (lane == 0) swv[wv] = lsum;
  __syncthreads();
  if (tid < 32) {
    float fs = (tid < 8) ? swv[tid] : 0.0f;
    for (int off = 4; off > 0; off >>= 1) fs += __shfl_xor(fs, off);
    if (tid == 0) swv[0] = fs;
  }
  __syncthreads();
  float Zbtm = swv[0];
  float Z = Zbtm + topk_expsum;
  float inv = 1.0f / Z;
  for (int s = tid; s < SS; s += 256) prow[s] = vals[s] * inv;
  if (tid == 0) Abtm[row] = Zbtm * inv;
}

// ---------------------------------------- WMMA GEMM: V_btm = P_btm * V
// P [NH,C,S], values [N,S,H,D] -> Vb [NH,C,D]  (Vb pre-zeroed)
// K-dim split SPLIT ways for occupancy; partial tiles accumulated with
// global float atomics.
__global__ void pv_gemm(const float* __restrict__ P,
                        const float* __restrict__ values,
                        float* __restrict__ Vb) {
  const int TPH = (CC / 16) * (DD / 16) * SPLIT;     // 256 waves per head
  int wid = blockIdx.x * (blockDim.x >> 5) + (threadIdx.x >> 5);
  int nh = wid / TPH;
  int t  = wid % TPH;
  int mt = t / ((DD / 16) * SPLIT);
  int t2 = t % ((DD / 16) * SPLIT);
  int nt = t2 / SPLIT;
  int sp = t2 % SPLIT;
  int n = nh / HH, h = nh % HH;
  int lane = threadIdx.x & 31;
  int half = lane >> 4, tl = lane & 15;
  int m0 = mt * 16, d0 = nt * 16, sbeg = sp * SCHUNK;

  const float* prow  = P + ((size_t)nh * CC + m0 + tl) * SS;
  const float* vbase = values + ((size_t)n * SS * HH + h) * DD + d0 + tl;
  const size_t vstride = (size_t)HH * DD;            // stride between key rows

  v8f acc = {};
  for (int s = sbeg; s < sbeg + SCHUNK; s += 4) {
    v2f a, b;
    a.x = prow[s + 2 * half];
    a.y = prow[s + 2 * half + 1];
    size_t r0 = (size_t)(s + 2 * half) * vstride;    // B[k][n] = V[s+k][d0+n]
    b.x = vbase[r0];
    b.y = vbase[r0 + vstride];
    __builtin_prefetch(vbase + r0 + 4 * vstride, 0, 1);  // next k-step rows
    acc = __builtin_amdgcn_wmma_f32_16x16x4_f32(
        false, a, false, b, (short)0, acc, false, false);
  }
  float* obase = Vb + ((size_t)nh * CC + m0 + 8 * half) * DD + d0 + tl;
#pragma unroll
  for (int r = 0; r < 8; ++r) atomicAdd(&obase[(size_t)r * DD], acc[r]);
}

// --------------------- per-query sparse top-k attention + combine (wave32)
__global__ void topk_attn(const float* __restrict__ queries,
                          const float* __restrict__ keys,
                          const float* __restrict__ values,
                          const int* __restrict__ cid,
                          const int* __restrict__ topk,
                          const float* __restrict__ Abtm,
                          const float* __restrict__ Vb,
                          float* __restrict__ out) {
  int wave = blockIdx.x * (blockDim.x >> 5) + (threadIdx.x >> 5); // query id
  int lane = threadIdx.x & 31;
  int l  = wave % LL;
  int nh = wave / LL;
  int h = nh % HH, n = nh / HH;
  int c = cid[(size_t)nh * LL + l];

  const float* q = queries + (((size_t)n * LL + l) * HH + h) * EE;
  int idx = topk[((size_t)nh * CC + c) * KK + lane];          // lane = k slot
  const float* krow = keys + (((size_t)n * SS + idx) * HH + h) * EE;

  float dot = 0.0f;
  for (int e = 0; e < EE; ++e) dot += q[e] * krow[e];
  float sc = TEMP * dot;

  float mx = sc;                                // wave32 max reduce
  for (int off = 16; off > 0; off >>= 1) mx = fmaxf(mx, __shfl_xor(mx, off));
  float ex = __expf(sc - mx);
  float sum = ex;
  for (int off = 16; off > 0; off >>= 1) sum += __shfl_xor(sum, off);

  float ab = Abtm[(size_t)nh * CC + c];
  float at = ex * (1.0f - ab) / sum;

  // each lane produces output dims d0, d0+1
  int d0 = 2 * lane;
  float acc0 = 0.0f, acc1 = 0.0f;
  for (int k = 0; k < KK; ++k) {
    float a = __shfl(at, k);
    int  id = __shfl(idx, k);
    const float* vrow = values + (((size_t)n * SS + id) * HH + h) * DD;
    acc0 += a * vrow[d0];
    acc1 += a * vrow[d0 + 1];
  }
  const float* vb = Vb + ((size_t)nh * CC + c) * DD;
  float* o = out + (((size_t)n * LL + l) * HH + h) * DD;
  o[d0]     = acc0 + vb[d0];
  o[d0 + 1] = acc1 + vb[d0 + 1];
}

// ---------------------------------------------------------------- launcher
extern "C" void kernel_launch(void* const* d_in, const int* in_sizes, int n_in,
                              void* d_out, int out_size, void* d_ws, size_t ws_size,
                              hipStream_t stream) {
  const float* queries = (const float*)d_in[0];  // [N,L,H,E]
  const float* keys    = (const float*)d_in[1];  // [N,S,H,E]
  const float* values  = (const float*)d_in[2];  // [N,S,H,D]
  const int*   cid     = (const int*)d_in[3];    // [N,H,L]
  float* out = (float*)d_out;                    // [N,L,H,D]

  float* ws   = (float*)d_ws;
  float* Qg   = ws;                    // NG  floats (sum, then mean)
  float* cnt  = Qg + NG;               // NHC floats
  float* P    = cnt + NHC;             // NP  floats (scores -> probs)
  int*   tki  = (int*)(P + NP);        // NTK ints
  float* Ab   = (float*)(tki + NTK);   // NHC floats
  float* Vb   = Ab + NHC;              // NVB floats

  // 1) zero Qsum + counts (contiguous), and Vb (atomic accumulation target)
  {
    size_t nz = (size_t)NG + NHC;
    zero_f32<<<(unsigned)((nz + 255) / 256), 256, 0, stream>>>(Qg, nz);
    zero_f32<<<(NVB + 255) / 256, 256, 0, stream>>>(Vb, (size_t)NVB);
  }
  // 2) cluster sums
  {
    size_t nq = (size_t)NN * HH * LL;
    cluster_accum<<<(unsigned)((nq + 255) / 256), 256, 0, stream>>>(queries, cid, Qg, cnt);
  }
  // 3) means
  finalize_means<<<(NG + 255) / 256, 256, 0, stream>>>(Qg, cnt);
  // 4) centroid GEMM: 16 heads * 512 strips, 8 waves/block
  {
    int waves = NN * HH * (CC / 16) * (SS / 64);
    qk_gemm<<<waves / 8, 256, 0, stream>>>(Qg, keys, P);
  }
  // 5) top-k + masked softmax: one block per (n,h,c) row
  topk_softmax<<<NN * HH * CC, 256, 0, stream>>>(P, tki, Ab);
  // 6) bottom-k GEMM with 4-way K split: 4096 waves
  {
    int waves = NN * HH * (CC / 16) * (DD / 16) * SPLIT;
    pv_gemm<<<waves / 8, 256, 0, stream>>>(P, values, Vb);
  }
  // 7) per-query sparse attention + combine: one wave per query
  {
    int waves = NN * HH * LL;
    topk_attn<<<waves / 8, 256, 0, stream>>>(queries, keys, values, cid, tki, Ab, Vb, out);
  }
}